// Attn_64381559767295
// MI455X (gfx1250) — compile-verified
//
#include <hip/hip_runtime.h>
#include <hip/hip_bf16.h>

// Problem constants (from the reference): B=8, S=8192, H=512
#define BB 8
#define SS 8192
#define HH 512
#define ROWS 4   // encoder rows per wave in the streaming kernel

typedef __attribute__((ext_vector_type(2))) float v2f;
typedef __attribute__((ext_vector_type(8))) float v8f;

// ---------------------------------------------------------------------------
// Kernel 1: v = hidden @ W  ([8,512], padded to M=16 for WMMA),
//           c[b] = dot(bias, hidden[b])
// 4 blocks x 8 waves = 32 waves; each wave owns one 16-column N-tile and runs
// a K-loop of 128 V_WMMA_F32_16X16X4_F32 steps.
//
// Padding trick: D row m depends only on A row m, and we discard D rows 8-15,
// so padding lanes load hid[(m&7)...] unconditionally -> no EXEC divergence,
// and the K-pair merges into a single b64 load.
// ---------------------------------------------------------------------------
__global__ __launch_bounds__(256) void proj_v_kernel(
    const float* __restrict__ hid,   // [B,H]
    const float* __restrict__ W,     // [H,H], W[k*H + n]
    const float* __restrict__ bias,  // [H]
    float* __restrict__ v_out,       // [B,H]
    float* __restrict__ c_out)       // [B]
{
    const int lane  = threadIdx.x & 31;
    const int wave  = threadIdx.x >> 5;            // 0..7
    const int g     = blockIdx.x * 8 + wave;       // 0..31 -> N-tile id
    const int nb    = g * 16;                      // N-tile base column
    const int m     = lane & 15;                   // tile row (A) / tile col (B)
    const int khalf = lane >> 4;                   // 0: K{0,1}, 1: K{2,3}

    // Per-lane base pointers (loop-invariant):
    //   A (16x4 f32): lane m, VGPR0/1 = K = 2*khalf + {0,1} within each k0 group
    //   B (4x16 f32): lane column m, rows mirror A's lane-half split
    const float* __restrict__ arow = hid + (m & 7) * HH + 2 * khalf;
    const float* __restrict__ bcol = W + (2 * khalf) * HH + nb + m;

    v8f acc = (v8f){0.f,0.f,0.f,0.f,0.f,0.f,0.f,0.f};

    #pragma unroll 4
    for (int k0 = 0; k0 < HH; k0 += 4) {
        v2f a;
        a.x = arow[k0 + 0];
        a.y = arow[k0 + 1];
        v2f bm;
        bm.x = bcol[(k0 + 0) * HH];
        bm.y = bcol[(k0 + 1) * HH];
        acc = __builtin_amdgcn_wmma_f32_16x16x4_f32(
            /*neg_a=*/false, a, /*neg_b=*/false, bm,
            /*c_mod=*/(short)0, acc, /*reuse_a=*/false, /*reuse_b=*/false);
    }

    // D layout: lanes 0-15 hold M=0..7 in VGPRs 0..7 (the 8 real batches), N=lane.
    if (lane < 16) {
        #pragma unroll
        for (int j = 0; j < BB; ++j)
            v_out[j * HH + nb + lane] = acc[j];
    }

    // c[b] = dot(bias, hidden[b]) — block 0 only, wave w handles batch w.
    if (blockIdx.x == 0) {
        float p = 0.0f;
        for (int k = lane; k < HH; k += 32)
            p += bias[k] * hid[wave * HH + k];
        #pragma unroll
        for (int off = 16; off > 0; off >>= 1)
            p += __shfl_xor(p, off, 32);
        if (lane == 0) c_out[wave] = p;
    }
}

// ---------------------------------------------------------------------------
// Kernel 2 (the 128 MB stream — bandwidth bound):
//   scores[b,s] = dot(enc[b,s,:], v[b,:]) + c[b]
// Each wave: preload v[b] into 4 float4 registers (amortized over ROWS rows),
// then per row 4 coalesced b128 loads/lane + FMAs + wave32 shuffle reduction.
// ---------------------------------------------------------------------------
__global__ __launch_bounds__(256) void score_kernel(
    const float* __restrict__ enc,   // [B,S,H]
    const float* __restrict__ v,     // [B,H]
    const float* __restrict__ c,     // [B]
    float* __restrict__ scores)      // [B,S]
{
    const int lane = threadIdx.x & 31;
    const int wave = threadIdx.x >> 5;                 // 0..7
    const int b    = blockIdx.y;
    const int s0   = (blockIdx.x * 8 + wave) * ROWS;

    const float4* __restrict__ vrow = (const float4*)(v + b * HH);
    float4 w4[4];
    #pragma unroll
    for (int i = 0; i < 4; ++i) w4[i] = vrow[i * 32 + lane];
    const float cb = c[b];

    const float4* __restrict__ erow =
        (const float4*)(enc + ((size_t)b * SS + s0) * (size_t)HH);

    // Speculative prefetch of this wave's next 4-row group (global_prefetch_b8).
    if (blockIdx.x + 1 < gridDim.x)
        __builtin_prefetch((const void*)(erow + 8 * ROWS * (HH / 4)), 0, 1);

    #pragma unroll
    for (int r = 0; r < ROWS; ++r) {
        const float4* __restrict__ p = erow + r * (HH / 4);
        float acc = 0.0f;
        #pragma unroll
        for (int i = 0; i < 4; ++i) {
            const float4 e = p[i * 32 + lane];
            acc += e.x * w4[i].x + e.y * w4[i].y + e.z * w4[i].z + e.w * w4[i].w;
        }
        #pragma unroll
        for (int off = 16; off > 0; off >>= 1)
            acc += __shfl_xor(acc, off, 32);
        if (lane == 0) scores[b * SS + s0 + r] = acc + cb;
    }
}

// ---------------------------------------------------------------------------
// Kernel 3: in-place softmax over S=8192 per batch. 1 block (32 waves) per batch.
// Each thread keeps its 8 elements in registers; all reads complete before the
// first barrier, so in-place on d_out is safe.
// ---------------------------------------------------------------------------
__global__ __launch_bounds__(1024) void softmax_kernel(float* __restrict__ data)
{
    const int b    = blockIdx.x;
    const int tid  = threadIdx.x;
    const int lane = tid & 31;
    const int wv   = tid >> 5;                   // 0..31
    float* row = data + b * SS;

    float vals[8];
    #pragma unroll
    for (int i = 0; i < 8; ++i) vals[i] = row[tid + i * 1024];

    __shared__ float red[32];

    // --- max reduction ---
    float m = vals[0];
    #pragma unroll
    for (int i = 1; i < 8; ++i) m = fmaxf(m, vals[i]);
    #pragma unroll
    for (int off = 16; off > 0; off >>= 1)
        m = fmaxf(m, __shfl_xor(m, off, 32));
    if (lane == 0) red[wv] = m;
    __syncthreads();
    if (wv == 0) {
        float mm = red[lane];
        #pragma unroll
        for (int off = 16; off > 0; off >>= 1)
            mm = fmaxf(mm, __shfl_xor(mm, off, 32));
        if (lane == 0) red[0] = mm;
    }
    __syncthreads();
    m = red[0];
    __syncthreads();

    // --- exp + sum reduction ---
    float sloc = 0.0f;
    #pragma unroll
    for (int i = 0; i < 8; ++i) { vals[i] = __expf(vals[i] - m); sloc += vals[i]; }
    #pragma unroll
    for (int off = 16; off > 0; off >>= 1)
        sloc += __shfl_xor(sloc, off, 32);
    if (lane == 0) red[wv] = sloc;
    __syncthreads();
    if (wv == 0) {
        float ss = red[lane];
        #pragma unroll
        for (int off = 16; off > 0; off >>= 1)
            ss += __shfl_xor(ss, off, 32);
        if (lane == 0) red[0] = ss;
    }
    __syncthreads();
    const float inv = 1.0f / red[0];

    #pragma unroll
    for (int i = 0; i < 8; ++i) row[tid + i * 1024] = vals[i] * inv;
}

// ---------------------------------------------------------------------------
extern "C" void kernel_launch(void* const* d_in, const int* in_sizes, int n_in,
                              void* d_out, int out_size, void* d_ws, size_t ws_size,
                              hipStream_t stream) {
    (void)in_sizes; (void)n_in; (void)out_size; (void)ws_size;
    const float* hid  = (const float*)d_in[0];   // [1,B,H]
    const float* enc  = (const float*)d_in[1];   // [B,S,H]
    const float* W    = (const float*)d_in[2];   // [H,H]
    const float* bias = (const float*)d_in[3];   // [H]
    float* out = (float*)d_out;                  // [B,S]

    float* v = (float*)d_ws;                     // [B,H] = 16 KB scratch
    float* c = v + BB * HH;                      // [B]

    proj_v_kernel<<<dim3(4), dim3(256), 0, stream>>>(hid, W, bias, v, c);
    score_kernel<<<dim3(SS / (8 * ROWS), BB), dim3(256), 0, stream>>>(enc, v, c, out);
    softmax_kernel<<<dim3(BB), dim3(1024), 0, stream>>>(out);
}